// EdgeBlock_45578192945250
// MI455X (gfx1250) — compile-verified
//
#include <hip/hip_runtime.h>
#include <hip/hip_bf16.h>

typedef __attribute__((ext_vector_type(2))) float v2f;
typedef __attribute__((ext_vector_type(8))) float v8f;

#define CCH 64
#define NEG_INF_I ((int)0xFF800000)

// ---------------------------------------------------------------------------
// Kernel 1: dual node-level GEMM via fp32 WMMA (V_WMMA_F32_16X16X4_F32).
//   P = x @ W1, Q = x @ W2   (W1 = W[0:64,:], W2 = W[64:128,:], row-major)
//   A_ws = P - Q + bias      (== x @ (W1-W2) + b)
//   B_ws = Q                 (== x @ W2)
// One wave per 16x16 tile; block = 4 waves covering the 4 column tiles.
// ---------------------------------------------------------------------------
__global__ __launch_bounds__(128) void k_gemm(const float* __restrict__ x,
                                              const float* __restrict__ W,
                                              const float* __restrict__ bias,
                                              float* __restrict__ A_ws,
                                              float* __restrict__ B_ws,
                                              int N)
{
    const int lane = threadIdx.x & 31;
    const int wave = threadIdx.x >> 5;
    const int n0   = wave * 16;
    const int m0   = blockIdx.x * 16;
    const int half = lane >> 4;      // 0: lanes 0-15, 1: lanes 16-31
    const int lm   = lane & 15;
    const int col  = n0 + lm;

    v8f accP = {};
    v8f accQ = {};

    int rowA = m0 + lm;
    if (rowA >= N) rowA = N - 1;     // clamp (N % 16 == 0 in practice)

#pragma unroll
    for (int kk = 0; kk < 16; ++kk) {
        const int kb  = kk * 4 + half * 2;     // this lane-half's K pair
        v2f a, b1, b2;
        // A fragment: 16x4, M = lane%16, K = kb, kb+1 across 2 VGPRs
        a.x  = x[(long)rowA * CCH + kb];
        a.y  = x[(long)rowA * CCH + kb + 1];
        // B fragments: 4x16, N = lane%16, K rows kb, kb+1
        b1.x = W[kb * CCH + col];
        b1.y = W[(kb + 1) * CCH + col];
        b2.x = W[(kb + 64) * CCH + col];
        b2.y = W[(kb + 65) * CCH + col];
        accP = __builtin_amdgcn_wmma_f32_16x16x4_f32(false, a, false, b1,
                                                     (short)0, accP, false, false);
        accQ = __builtin_amdgcn_wmma_f32_16x16x4_f32(false, a, false, b2,
                                                     (short)0, accQ, false, false);
    }

    const float bv = bias[col];
    if (m0 + 16 <= N) {
        // Fast path: whole tile in-bounds (always taken when N % 16 == 0).
#pragma unroll
        for (int r = 0; r < 8; ++r) {
            const int row = m0 + r + half * 8;   // D layout: M = r + 8*half
            const float q = accQ[r];
            A_ws[(long)row * CCH + col] = accP[r] - q + bv;
            B_ws[(long)row * CCH + col] = q;
        }
    } else {
#pragma unroll
        for (int r = 0; r < 8; ++r) {
            const int row = m0 + r + half * 8;
            if (row < N) {
                const float q = accQ[r];
                A_ws[(long)row * CCH + col] = accP[r] - q + bv;
                B_ws[(long)row * CCH + col] = q;
            }
        }
    }
}

// ---------------------------------------------------------------------------
// Kernel 2: init scatter-max buffer to -inf (bit pattern), stored as int.
// ---------------------------------------------------------------------------
__global__ void k_init(int* __restrict__ M, long total)
{
    long i = (long)blockIdx.x * blockDim.x + threadIdx.x;
    if (i < total) M[i] = NEG_INF_I;
}

// ---------------------------------------------------------------------------
// Kernel 3: per-edge scatter-max.  M[dst][c] = max(M[dst][c], B[src][c]).
// Float max via sign-split integer atomics (native global_atomic_max_i32 /
// min_u32, order-independent => deterministic). M buffer (25.6MB) lives in L2.
// ---------------------------------------------------------------------------
__global__ __launch_bounds__(256) void k_scatter(const float* __restrict__ Bn,
                                                 const int* __restrict__ ei,
                                                 int* __restrict__ M,
                                                 int E)
{
    long tid   = (long)blockIdx.x * blockDim.x + threadIdx.x;
    long total = (long)E * CCH;
    if (tid >= total) return;
    int e = (int)(tid >> 6);
    int c = (int)(tid & 63);
    int src = ei[e];
    int dst = ei[E + e];
    float v = Bn[(long)src * CCH + c];
    int vi = __float_as_int(v);
    int* addr = M + (long)dst * CCH + c;
    if (vi >= 0) atomicMax(addr, vi);
    else         atomicMin((unsigned int*)addr, (unsigned int)vi);
}

// ---------------------------------------------------------------------------
// Kernel 4: h = ELU(A + M) (0 where no edges); write h to out, and emit
// deterministic per-block partial sums / sums-of-squares for BatchNorm.
// Each block handles 64 node rows; thread = (row%4 strip, channel).
// ---------------------------------------------------------------------------
__global__ __launch_bounds__(256) void k_elu_partial(const float* __restrict__ A_ws,
                                                     const int* __restrict__ M,
                                                     float* __restrict__ H,
                                                     float* __restrict__ psum,
                                                     float* __restrict__ psq,
                                                     int N)
{
    __shared__ float smem[256];
    const int t  = threadIdx.x;
    const int c  = t & 63;
    const int r4 = t >> 6;
    const int base = blockIdx.x * 64;

    float s = 0.f, q = 0.f;
    for (int rr = r4; rr < 64; rr += 4) {
        int row = base + rr;
        if (row < N) {
            long idx = (long)row * CCH + c;
            int  mb  = M[idx];
            float h;
            if (mb == NEG_INF_I) {
                h = 0.f;                       // empty node: -inf -> 0, ELU(0)=0
            } else {
                float z = A_ws[idx] + __int_as_float(mb);
                h = (z > 0.f) ? z : expm1f(z); // ELU, alpha=1
            }
            H[idx] = h;
            s += h;
            q += h * h;
        }
    }
    smem[t] = s; __syncthreads();
    if (r4 == 0)
        psum[(long)blockIdx.x * 64 + c] = smem[c] + smem[c + 64] + smem[c + 128] + smem[c + 192];
    __syncthreads();
    smem[t] = q; __syncthreads();
    if (r4 == 0)
        psq[(long)blockIdx.x * 64 + c]  = smem[c] + smem[c + 64] + smem[c + 128] + smem[c + 192];
}

// ---------------------------------------------------------------------------
// Kernel 5: combine partials (fixed order, double accum) -> mean, 1/sqrt(var+eps)
// ---------------------------------------------------------------------------
__global__ void k_stats(const float* __restrict__ psum,
                        const float* __restrict__ psq,
                        int nblocks, int N,
                        float* __restrict__ mean,
                        float* __restrict__ invstd)
{
    int c = threadIdx.x;
    if (c >= 64) return;
    double s = 0.0, q = 0.0;
    for (int b = 0; b < nblocks; ++b) {
        s += (double)psum[(long)b * 64 + c];
        q += (double)psq[(long)b * 64 + c];
    }
    double m = s / (double)N;
    double v = q / (double)N - m * m;   // biased variance
    if (v < 0.0) v = 0.0;
    mean[c]   = (float)m;
    invstd[c] = (float)(1.0 / sqrt(v + 1e-5));
}

// ---------------------------------------------------------------------------
// Kernel 6: in-place BatchNorm affine on the staged h values in d_out.
// ---------------------------------------------------------------------------
__global__ __launch_bounds__(256) void k_bn(float* __restrict__ out,
                                            const float* __restrict__ mean,
                                            const float* __restrict__ invstd,
                                            const float* __restrict__ gamma,
                                            const float* __restrict__ beta,
                                            long total)
{
    long i = (long)blockIdx.x * blockDim.x + threadIdx.x;
    if (i < total) {
        int c = (int)(i & 63);
        out[i] = gamma[c] * (out[i] - mean[c]) * invstd[c] + beta[c];
    }
}

// ---------------------------------------------------------------------------
extern "C" void kernel_launch(void* const* d_in, const int* in_sizes, int n_in,
                              void* d_out, int out_size, void* d_ws, size_t ws_size,
                              hipStream_t stream)
{
    const float* x     = (const float*)d_in[0];
    const int*   ei    = (const int*)  d_in[1];
    const float* W     = (const float*)d_in[2];
    const float* bias  = (const float*)d_in[3];
    const float* gamma = (const float*)d_in[4];
    const float* beta  = (const float*)d_in[5];
    float* out = (float*)d_out;

    const int N = in_sizes[0] / CCH;   // 100000
    const int E = in_sizes[1] / 2;     // 1600000

    // Workspace carve-up (all float-sized): A | B | M(int) | psum | psq | stats
    const size_t nodef = (size_t)N * CCH;
    float* A_ws = (float*)d_ws;
    float* B_ws = A_ws + nodef;
    int*   M_ws = (int*)(B_ws + nodef);
    const int nb4 = (N + 63) / 64;
    float* psum    = (float*)(M_ws + nodef);
    float* psq     = psum + (size_t)nb4 * 64;
    float* meanp   = psq  + (size_t)nb4 * 64;
    float* invstdp = meanp + 64;

    const long tot  = (long)nodef;
    const long etot = (long)E * CCH;

    k_gemm<<<(N + 15) / 16, 128, 0, stream>>>(x, W, bias, A_ws, B_ws, N);
    k_init<<<(int)((tot + 255) / 256), 256, 0, stream>>>(M_ws, tot);
    k_scatter<<<(int)((etot + 255) / 256), 256, 0, stream>>>(B_ws, ei, M_ws, E);
    k_elu_partial<<<nb4, 256, 0, stream>>>(A_ws, M_ws, out, psum, psq, N);
    k_stats<<<1, 64, 0, stream>>>(psum, psq, nb4, N, meanp, invstdp);
    k_bn<<<(int)((tot + 255) / 256), 256, 0, stream>>>(out, meanp, invstdp, gamma, beta, tot);
}